// SimpleGAT_70695161692282
// MI455X (gfx1250) — compile-verified
//
#include <hip/hip_runtime.h>
#include <math.h>

typedef __attribute__((ext_vector_type(2))) float v2f;
typedef __attribute__((ext_vector_type(8))) float v8f;

#define NEG_SLOPE 0.2f
#define GEMM_BS   256
#define GEMM_WAVES 8

// ---------------------------------------------------------------------------
// GEMM: D[M,Nc] = A[M,K] @ B[K,Nc], f32 via V_WMMA_F32_16X16X4_F32.
// 8 waves per block; A strip staged through LDS with coalesced b128 loads,
// WMMA A-fragments then read from LDS (ds path). K mult of 4, Nc mult of 16,
// K <= 128, rowsPerBlock = 128*16/Nc (16 for Nc=128, 64 for Nc=32).
// ---------------------------------------------------------------------------
__global__ void gat_gemm_wmma_f32(const float* __restrict__ A,
                                  const float* __restrict__ B,
                                  float* __restrict__ D,
                                  int M, int K, int Nc) {
  __shared__ float sA[64 * 128];                  // 32 KB max strip

  int tilesN       = Nc >> 4;                     // 8 or 2
  int rowsPerBlock = (GEMM_WAVES / tilesN) << 4;  // 16 or 64
  int blockRow0    = blockIdx.x * rowsPerBlock;

  // Cooperative, coalesced stage of the A strip [rowsPerBlock x K]
  int total4 = (rowsPerBlock * K) >> 2;
  for (int i = threadIdx.x; i < total4; i += blockDim.x) {
    int idx  = i << 2;
    int rr   = idx / K;
    int cc   = idx - rr * K;
    int grow = blockRow0 + rr;
    float4 v = (grow < M) ? *(const float4*)(A + (size_t)grow * K + cc)
                          : make_float4(0.f, 0.f, 0.f, 0.f);
    *(float4*)(sA + idx) = v;
  }
  __syncthreads();

  int wave  = threadIdx.x >> 5;
  int lane  = threadIdx.x & 31;
  int colT  = wave % tilesN;
  int rowG  = wave / tilesN;
  int lrow0 = rowG << 4;
  int grow0 = blockRow0 + lrow0;
  if (grow0 >= M) return;        // wave-uniform: EXEC all-ones inside WMMA

  int r    = lane & 15;
  int half = lane >> 4;
  int col0 = colT << 4;

  v8f acc = {};
  const float* sar = sA + (lrow0 + r) * K;
  for (int k = 0; k < K; k += 4) {
    int kk = k + 2 * half;
    v2f a, b;
    a[0] = sar[kk];                               // ds_load from LDS
    a[1] = sar[kk + 1];
    b[0] = B[(size_t)kk * Nc + col0 + r];         // W is WGP$/L2 resident
    b[1] = B[(size_t)(kk + 1) * Nc + col0 + r];
    // 8 args: (neg_a, A, neg_b, B, c_mod, C, reuse_a, reuse_b)
    acc = __builtin_amdgcn_wmma_f32_16x16x4_f32(false, a, false, b,
                                                (short)0, acc, false, false);
  }
#pragma unroll
  for (int g = 0; g < 8; ++g)
    D[(size_t)(grow0 + g + 8 * half) * Nc + col0 + r] = acc[g];
}

// ---------------------------------------------------------------------------
// Per-node attention scores: s_src[n,h] = dot(h[n,h,:], a_src[h,:]) etc.
// ---------------------------------------------------------------------------
__global__ void gat_scores(const float* __restrict__ hfeat,
                           const float* __restrict__ a_src,
                           const float* __restrict__ a_dst,
                           float* __restrict__ s_src,
                           float* __restrict__ s_dst,
                           long long N, int H, int C) {
  long long t = (long long)blockIdx.x * blockDim.x + threadIdx.x;
  if (t >= N * H) return;
  long long n = t / H;
  int hh = (int)(t - n * H);
  const float* hp = hfeat + (n * H + hh) * (long long)C;
  const float* as = a_src + (long long)hh * C;
  const float* ad = a_dst + (long long)hh * C;
  float ss = 0.f, sd = 0.f;
#pragma unroll 8
  for (int c = 0; c < C; ++c) {
    float v = hp[c];
    ss = fmaf(v, as[c], ss);
    sd = fmaf(v, ad[c], sd);
  }
  s_src[t] = ss;
  s_dst[t] = sd;
}

// ---------------------------------------------------------------------------
// Helpers
// ---------------------------------------------------------------------------
__device__ __forceinline__ void edge_endpoints(const int* __restrict__ ei,
                                               long long E, long long eid,
                                               int& s, int& d) {
  if (eid < E) { s = ei[eid]; d = ei[E + eid]; }
  else         { s = d = (int)(eid - E); }      // self-loops appended
}

__device__ __forceinline__ void atomicMaxF(float* addr, float val) {
  if (val >= 0.f) atomicMax((int*)addr, __float_as_int(val));
  else            atomicMin((unsigned int*)addr, __float_as_uint(val));
}

__global__ void gat_fill(float* __restrict__ p, float v, long long n) {
  long long t = (long long)blockIdx.x * blockDim.x + threadIdx.x;
  if (t < n) p[t] = v;
}

// ---------------------------------------------------------------------------
// Edge pass 1: segment max of leaky_relu(s_src[src]+s_dst[dst]) over dst
// ---------------------------------------------------------------------------
__global__ void gat_edge_max(const int* __restrict__ ei, long long E,
                             long long Etot, int H,
                             const float* __restrict__ s_src,
                             const float* __restrict__ s_dst,
                             float* __restrict__ emax) {
  long long t = (long long)blockIdx.x * blockDim.x + threadIdx.x;
  if (t >= Etot * H) return;
  long long eid = t / H;
  int hh = (int)(t - eid * H);
  int s, d; edge_endpoints(ei, E, eid, s, d);
  float e = s_src[(long long)s * H + hh] + s_dst[(long long)d * H + hh];
  e = e > 0.f ? e : NEG_SLOPE * e;
  atomicMaxF(&emax[(long long)d * H + hh], e);
}

// ---------------------------------------------------------------------------
// Edge pass 2: e_exp = exp(e - emax[dst]); store it, segment-sum into denom
// ---------------------------------------------------------------------------
__global__ void gat_edge_expsum(const int* __restrict__ ei, long long E,
                                long long Etot, int H,
                                const float* __restrict__ s_src,
                                const float* __restrict__ s_dst,
                                const float* __restrict__ emax,
                                float* __restrict__ eexp,
                                float* __restrict__ denom) {
  long long t = (long long)blockIdx.x * blockDim.x + threadIdx.x;
  if (t >= Etot * H) return;
  long long eid = t / H;
  int hh = (int)(t - eid * H);
  int s, d; edge_endpoints(ei, E, eid, s, d);
  float e = s_src[(long long)s * H + hh] + s_dst[(long long)d * H + hh];
  e = e > 0.f ? e : NEG_SLOPE * e;
  float ex = expf(e - emax[(long long)d * H + hh]);
  eexp[t] = ex;
  atomicAdd(&denom[(long long)d * H + hh], ex);
}

// ---------------------------------------------------------------------------
// Edge pass 3: out[dst,h,:] += alpha * h[src,h,:]  (float4 vectorized)
// One thread handles (edge, head, 4 channels).
// ---------------------------------------------------------------------------
__global__ void gat_edge_aggr(const int* __restrict__ ei, long long E,
                              long long Etot, int H, int C,
                              const float* __restrict__ hfeat,
                              const float* __restrict__ eexp,
                              const float* __restrict__ denom,
                              float* __restrict__ outf) {
  int vecs = C >> 2;
  long long t = (long long)blockIdx.x * blockDim.x + threadIdx.x;
  long long total = Etot * H * vecs;
  if (t >= total) return;
  long long t1 = t / vecs;
  int v = (int)(t - t1 * vecs);
  long long eid = t1 / H;
  int hh = (int)(t1 - eid * H);
  int s, d; edge_endpoints(ei, E, eid, s, d);
  float al = eexp[t1] / (denom[(long long)d * H + hh] + 1e-16f);
  const float4 hv = *((const float4*)(hfeat + ((long long)s * H + hh) * C) + v);
  float* op = outf + ((long long)d * H + hh) * C + (v << 2);
  atomicAdd(op + 0, hv.x * al);
  atomicAdd(op + 1, hv.y * al);
  atomicAdd(op + 2, hv.z * al);
  atomicAdd(op + 3, hv.w * al);
}

// ---------------------------------------------------------------------------
// Elementwise epilogues
// ---------------------------------------------------------------------------
__global__ void gat_bias_elu(float* __restrict__ x, const float* __restrict__ b,
                             long long n, int F) {
  long long t = (long long)blockIdx.x * blockDim.x + threadIdx.x;
  if (t >= n) return;
  float v = x[t] + b[t % F];
  x[t] = v > 0.f ? v : expf(v) - 1.f;
}

__global__ void gat_bias_add(float* __restrict__ x, const float* __restrict__ b,
                             long long n, int F) {
  long long t = (long long)blockIdx.x * blockDim.x + threadIdx.x;
  if (t >= n) return;
  x[t] = x[t] + b[t % F];
}

// ---------------------------------------------------------------------------
// Host-side orchestration
// ---------------------------------------------------------------------------
static inline unsigned nblk(long long n, int bs = 256) {
  return (unsigned)((n + bs - 1) / bs);
}

extern "C" void kernel_launch(void* const* d_in, const int* in_sizes, int n_in,
                              void* d_out, int out_size, void* d_ws, size_t ws_size,
                              hipStream_t stream) {
  const float* x      = (const float*)d_in[0];
  const int*   ei     = (const int*)d_in[1];
  const float* W1     = (const float*)d_in[2];
  const float* a_src1 = (const float*)d_in[3];
  const float* a_dst1 = (const float*)d_in[4];
  const float* b1     = (const float*)d_in[5];
  const float* W2     = (const float*)d_in[6];
  const float* a_src2 = (const float*)d_in[7];
  const float* a_dst2 = (const float*)d_in[8];
  const float* b2     = (const float*)d_in[9];
  float* out = (float*)d_out;

  const int Fin = 128, H1 = 4, C1 = 32, F1 = H1 * C1;   // 128
  const int H2 = 1, C2 = 32;
  long long N    = in_sizes[0] / Fin;     // 100000
  long long E    = in_sizes[1] / 2;       // 1.6M
  long long Etot = E + N;                 // + self loops

  // Workspace carve-up (floats)
  float* ws    = (float*)d_ws;
  float* hlin1 = ws;            ws += N * F1;      // layer-1 projected features
  float* out1  = ws;            ws += N * F1;      // layer-1 aggregation -> h1
  float* ssrc1 = ws;            ws += N * H1;
  float* sdst1 = ws;            ws += N * H1;
  float* emax1 = ws;            ws += N * H1;
  float* den1  = ws;            ws += N * H1;
  float* eexp  = ws;            ws += Etot * H1;   // reused by layer 2 (needs Etot*1)
  float* hlin2 = ws;            ws += N * C2;
  float* ssrc2 = ws;            ws += N;
  float* sdst2 = ws;            ws += N;
  float* emax2 = ws;            ws += N;
  float* den2  = ws;            ws += N;
  (void)ws_size; (void)n_in; (void)out_size;

  const float NEGINF = -INFINITY;

  // ---------------- Layer 1 ----------------
  {
    int rowsPerBlock = 128 * 16 / F1;   // 16
    unsigned blocks = (unsigned)((N + rowsPerBlock - 1) / rowsPerBlock);
    gat_gemm_wmma_f32<<<blocks, GEMM_BS, 0, stream>>>(x, W1, hlin1,
                                                      (int)N, Fin, F1);
  }
  gat_scores<<<nblk(N * H1), 256, 0, stream>>>(hlin1, a_src1, a_dst1,
                                               ssrc1, sdst1, N, H1, C1);
  gat_fill<<<nblk(N * H1), 256, 0, stream>>>(emax1, NEGINF, N * H1);
  gat_fill<<<nblk(N * H1), 256, 0, stream>>>(den1, 0.f, N * H1);
  gat_fill<<<nblk(N * F1), 256, 0, stream>>>(out1, 0.f, N * F1);

  gat_edge_max<<<nblk(Etot * H1), 256, 0, stream>>>(ei, E, Etot, H1,
                                                    ssrc1, sdst1, emax1);
  gat_edge_expsum<<<nblk(Etot * H1), 256, 0, stream>>>(ei, E, Etot, H1,
                                                       ssrc1, sdst1, emax1,
                                                       eexp, den1);
  gat_edge_aggr<<<nblk(Etot * H1 * (C1 / 4)), 256, 0, stream>>>(
      ei, E, Etot, H1, C1, hlin1, eexp, den1, out1);

  gat_bias_elu<<<nblk(N * F1), 256, 0, stream>>>(out1, b1, N * F1, F1);

  // ---------------- Layer 2 ----------------
  {
    int rowsPerBlock = 128 * 16 / C2;   // 64
    unsigned blocks = (unsigned)((N + rowsPerBlock - 1) / rowsPerBlock);
    gat_gemm_wmma_f32<<<blocks, GEMM_BS, 0, stream>>>(out1, W2, hlin2,
                                                      (int)N, F1, C2);
  }
  gat_scores<<<nblk(N * H2), 256, 0, stream>>>(hlin2, a_src2, a_dst2,
                                               ssrc2, sdst2, N, H2, C2);
  gat_fill<<<nblk(N), 256, 0, stream>>>(emax2, NEGINF, N);
  gat_fill<<<nblk(N), 256, 0, stream>>>(den2, 0.f, N);
  gat_fill<<<nblk(N * C2), 256, 0, stream>>>(out, 0.f, N * C2);

  gat_edge_max<<<nblk(Etot * H2), 256, 0, stream>>>(ei, E, Etot, H2,
                                                    ssrc2, sdst2, emax2);
  gat_edge_expsum<<<nblk(Etot * H2), 256, 0, stream>>>(ei, E, Etot, H2,
                                                       ssrc2, sdst2, emax2,
                                                       eexp, den2);
  gat_edge_aggr<<<nblk(Etot * H2 * (C2 / 4)), 256, 0, stream>>>(
      ei, E, Etot, H2, C2, hlin2, eexp, den2, out);

  // mean over 1 head is identity; add b2
  gat_bias_add<<<nblk(N * C2), 256, 0, stream>>>(out, b2, N * C2, C2);
}